// SimpleGCN_71339406787240
// MI455X (gfx1250) — compile-verified
//
#include <hip/hip_runtime.h>

typedef __attribute__((ext_vector_type(2))) float v2f;
typedef __attribute__((ext_vector_type(8))) float v8f;

// ---------------- degree / norm kernels ----------------
__global__ void k_init_deg(float* __restrict__ dout, float* __restrict__ din, int N) {
  int i = blockIdx.x * blockDim.x + threadIdx.x;
  if (i < N) { dout[i] = 1.0f; din[i] = 1.0f; }  // self-loop contributes 1 to each
}

__global__ void k_count_deg(const int* __restrict__ src, const int* __restrict__ dst,
                            float* __restrict__ dout, float* __restrict__ din, int E) {
  int e = blockIdx.x * blockDim.x + threadIdx.x;
  if (e < E) {
    unsafeAtomicAdd(&dout[src[e]], 1.0f);   // global_atomic_add_f32, no return
    unsafeAtomicAdd(&din[dst[e]], 1.0f);
  }
}

__global__ void k_deg_to_norm(float* __restrict__ dout, float* __restrict__ din, int N) {
  int i = blockIdx.x * blockDim.x + threadIdx.x;
  if (i < N) { dout[i] = rsqrtf(dout[i]); din[i] = rsqrtf(din[i]); }
}

// ---------------- WMMA GEMM: H = (X @ W) * norm_src[:,None]; AGG = H ----------------
// X: [N,K] row-major, W: [K,64] row-major. Each wave computes one 16x16 tile via
// v_wmma_f32_16x16x4_f32. Block = 256 thr = 8 waves covering 32 rows x 64 cols.
template <int K>
__global__ __launch_bounds__(256) void k_gemm_scale(
    const float* __restrict__ X, const float* __restrict__ W,
    const float* __restrict__ norm_src,
    float* __restrict__ H, float* __restrict__ AGG, int N) {
  constexpr int LDSW = 80;  // padded row stride: disjoint bank sets for the two half-waves
  __shared__ float Wl[K * LDSW];

  const int tid = threadIdx.x;
  for (int i = tid; i < K * 64; i += 256) {
    int k = i >> 6, c = i & 63;
    Wl[k * LDSW + c] = W[i];
  }
  __syncthreads();

  const int wave = tid >> 5;
  const int lane = tid & 31;
  const int rt = wave >> 2;            // 0..1 row tile
  const int ct = wave & 3;             // 0..3 col tile
  const int row0 = blockIdx.x * 32 + rt * 16;
  const int col0 = ct * 16;
  const int m  = lane & 15;
  const int hi = lane >> 4;            // lanes 16-31 hold K+2/K+3 (A) per ISA layout

  int rA = row0 + m; if (rA >= N) rA = N - 1;   // read clamp (N % 32 == 0 here anyway)
  const float* xrow = X + (size_t)rA * K;

  v8f acc = {};
#pragma unroll
  for (int k0 = 0; k0 < K; k0 += 4) {
    const int ka = k0 + 2 * hi;
    // A 16x4 frag: lanes 0-15 -> {K=k0,k0+1}, lanes 16-31 -> {k0+2,k0+3}
    v2f a = *(const v2f*)(xrow + ka);                 // 8B-aligned global_load_b64
    // B 4x16 frag mirrors A: rows ka, ka+1 at columns col0+m
    v2f b;
    b.x = Wl[ka * LDSW + col0 + m];
    b.y = Wl[(ka + 1) * LDSW + col0 + m];
    acc = __builtin_amdgcn_wmma_f32_16x16x4_f32(
        /*neg_a=*/false, a, /*neg_b=*/false, b,
        /*c_mod=*/(short)0, acc, /*reuse_a=*/false, /*reuse_b=*/false);
  }

  // All waves finish reading X before any stores -> layer 2 may run with AGG==X in place.
  __syncthreads();

  // Branch-free epilogue when the whole 16-row tile is in range (always true for
  // N % 32 == 0, i.e. this problem's N=100000); guarded fallback otherwise.
  const int col = col0 + m;
  const int rbase = row0 + hi * 8;     // C/D layout: VGPR r -> M = r (+8 for hi lanes)
  if (row0 + 16 <= N) {
#pragma unroll
    for (int r = 0; r < 8; ++r) {
      const int row = rbase + r;
      const float v = acc[r] * norm_src[row];
      const size_t o = (size_t)row * 64 + col;
      H[o] = v;     // scaled message
      AGG[o] = v;   // self-loop contribution pre-seeded (src==dst==row)
    }
  } else {
#pragma unroll
    for (int r = 0; r < 8; ++r) {
      const int row = rbase + r;
      if (row < N) {
        const float v = acc[r] * norm_src[row];
        const size_t o = (size_t)row * 64 + col;
        H[o] = v;
        AGG[o] = v;
      }
    }
  }
}

// ---------------- edge scatter: AGG[dst] += H[src] ----------------
// 16 threads per edge, float4 each: fully coalesced 256B gather per edge,
// 4 non-returning f32 atomics per thread resolving in the 192MB L2.
__global__ void k_scatter(const float* __restrict__ H, const int* __restrict__ src,
                          const int* __restrict__ dst, float* __restrict__ AGG, int E) {
  long long t = (long long)blockIdx.x * blockDim.x + threadIdx.x;
  int e = (int)(t >> 4);
  if (e >= E) return;
  int g = ((int)t & 15) << 2;
  int s = src[e], d = dst[e];
  float4 v = *(const float4*)(H + (size_t)s * 64 + g);   // global_load_b128
  float* ap = AGG + (size_t)d * 64 + g;
  unsafeAtomicAdd(ap + 0, v.x);
  unsafeAtomicAdd(ap + 1, v.y);
  unsafeAtomicAdd(ap + 2, v.z);
  unsafeAtomicAdd(ap + 3, v.w);
}

// ---------------- finalize: OUT = [relu](AGG * norm_dst[:,None] + b) ----------------
__global__ void k_finalize(const float* __restrict__ AGG, const float* __restrict__ norm_dst,
                           const float* __restrict__ b, float* __restrict__ OUT,
                           int N, int relu) {
  int t = blockIdx.x * blockDim.x + threadIdx.x;
  if (t >= N * 64) return;
  int i = t >> 6, c = t & 63;
  float v = fmaf(AGG[t], norm_dst[i], b[c]);
  if (relu) v = fmaxf(v, 0.0f);
  OUT[t] = v;
}

extern "C" void kernel_launch(void* const* d_in, const int* in_sizes, int n_in,
                              void* d_out, int out_size, void* d_ws, size_t ws_size,
                              hipStream_t stream) {
  const float* feat = (const float*)d_in[0];
  const int*   src  = (const int*)d_in[1];
  const int*   dst  = (const int*)d_in[2];
  const float* W1   = (const float*)d_in[3];
  const float* b1   = (const float*)d_in[4];
  const float* W2   = (const float*)d_in[5];
  const float* b2   = (const float*)d_in[6];
  float* out = (float*)d_out;

  const int N = in_sizes[0] / 32;
  const int E = in_sizes[1];

  // Workspace layout (floats): norm_src[N] | norm_dst[N] | H[N*64] | AGG[N*64]  (~52 MB)
  float* ws       = (float*)d_ws;
  float* norm_src = ws;
  float* norm_dst = ws + N;
  float* H        = ws + 2 * (size_t)N;
  float* AGG      = H + (size_t)N * 64;

  const int nbN  = (N + 255) / 256;
  const int nbE  = (E + 255) / 256;
  const int nbNF = (N * 64 + 255) / 256;
  const int nbG  = (N + 31) / 32;
  const long long scatter_threads = (long long)E * 16;
  const int nbS  = (int)((scatter_threads + 255) / 256);

  // degrees -> norms
  k_init_deg<<<nbN, 256, 0, stream>>>(norm_src, norm_dst, N);
  k_count_deg<<<nbE, 256, 0, stream>>>(src, dst, norm_src, norm_dst, E);
  k_deg_to_norm<<<nbN, 256, 0, stream>>>(norm_src, norm_dst, N);

  // layer 1: h1 = (feat@W1)*norm_src ; agg1 = h1 (self-loop) ; scatter ; relu-finalize (in place)
  k_gemm_scale<32><<<nbG, 256, 0, stream>>>(feat, W1, norm_src, H, AGG, N);
  k_scatter<<<nbS, 256, 0, stream>>>(H, src, dst, AGG, E);
  k_finalize<<<nbNF, 256, 0, stream>>>(AGG, norm_dst, b1, AGG, N, /*relu=*/1);

  // layer 2: x1 lives in AGG; GEMM reads/writes AGG in place (barrier-protected in kernel)
  k_gemm_scale<64><<<nbG, 256, 0, stream>>>(AGG, W2, norm_src, H, AGG, N);
  k_scatter<<<nbS, 256, 0, stream>>>(H, src, dst, AGG, E);
  k_finalize<<<nbNF, 256, 0, stream>>>(AGG, norm_dst, b2, out, N, /*relu=*/0);
}